// CosinLoss_33732673143498
// MI455X (gfx1250) — compile-verified
//
#include <hip/hip_runtime.h>
#include <hip/hip_bf16.h>

// CosinLoss on MI455X (gfx1250, wave32).
// Memory-bound: 128 MiB streamed @ 23.3 TB/s -> ~5.6 us floor.
// Row-dot (diag of P A^T) via V_WMMA_F32_16X16X4_F32; norms via VALU FMAs on
// the same loaded registers; split-K x4 per block for occupancy (4096 waves).

typedef float v2f __attribute__((ext_vector_type(2)));
typedef float v8f __attribute__((ext_vector_type(8)));

#define TILE_M 16
#define WAVES_PER_BLOCK 4
#define EPSN 1e-12f

__global__ __launch_bounds__(128)
void CosinLoss_wmma_kernel(const float* __restrict__ pc,
                           const float* __restrict__ aug,
                           float* __restrict__ out, int N, int D) {
  __shared__ float s_pp[TILE_M];
  __shared__ float s_aa[TILE_M];
  __shared__ float s_pa[TILE_M];

  const int lane = threadIdx.x & 31;
  const int wave = threadIdx.x >> 5;          // 0..3, owns a K-segment
  const int rb   = blockIdx.x * TILE_M;       // first row of this 16-row tile

  if (threadIdx.x < TILE_M) {
    s_pp[threadIdx.x] = 0.f;
    s_aa[threadIdx.x] = 0.f;
    s_pa[threadIdx.x] = 0.f;
  }
  __syncthreads();

  const int seg   = D / WAVES_PER_BLOCK;      // 256 for D=1024
  const int ks    = wave * seg;               // this wave's K-segment start
  const int steps = seg >> 2;                 // K=4 consumed per WMMA step

  // f32 WMMA A(16x4) layout: lane = m + 16*(k>=2), vgpr = k%2.
  // B(4x16) layout: lane = n + 16*(k>=2), vgpr = k%2.
  // => loading X rows as A == loading X^T as B with identical registers:
  //    each lane grabs a float2 at row (rb + lane%16), col (ks + 4k + 2*(lane>>4)).
  const int  m    = lane & 15;
  const int  hi   = lane >> 4;
  const long base = (long)(rb + m) * (long)D + (long)(ks + 2 * hi);
  const float* pptr = pc  + base;
  const float* aptr = aug + base;

  v8f   acc = {};          // Gram tile accumulator (pa on the diagonal)
  float pp  = 0.f;         // lane-partial sum of pc^2
  float aa  = 0.f;         // lane-partial sum of aug^2

  #pragma unroll 8
  for (int k = 0; k < steps; ++k) {
    v2f a = *(const v2f*)(pptr + 4 * k);
    v2f b = *(const v2f*)(aptr + 4 * k);
    pp = fmaf(a[0], a[0], fmaf(a[1], a[1], pp));
    aa = fmaf(b[0], b[0], fmaf(b[1], b[1], aa));
    // D(16x16) += A(16x4) x B(4x16): diag accumulates sum_k pc[m,k]*aug[m,k]
    acc = __builtin_amdgcn_wmma_f32_16x16x4_f32(
        /*neg_a=*/false, a, /*neg_b=*/false, b,
        /*c_mod=*/(short)0, acc, /*reuse_a=*/false, /*reuse_b=*/false);
  }

  // Lanes m and m+16 hold the same row -> fold the K halves of pp/aa.
  pp += __shfl_xor(pp, 16, 32);
  aa += __shfl_xor(aa, 16, 32);

  // Diagonal of C/D layout: (m,m) at lane=m,vgpr=m (m<8) or lane=m+16,vgpr=m-8.
  float pa = 0.f;
  #pragma unroll
  for (int i = 0; i < 8; ++i) {
    if (lane == i || lane == 24 + i) pa = acc[i];
  }
  const bool active = (lane < 8) || (lane >= 24);  // one lane per row m
  if (active) {
    atomicAdd(&s_pa[m], pa);   // ds_add_f32: combine the 4 K-segments
    atomicAdd(&s_pp[m], pp);
    atomicAdd(&s_aa[m], aa);
  }
  __syncthreads();

  if (wave == 0) {
    float v = 0.f;
    if (lane < TILE_M) {
      float np = fmaxf(sqrtf(s_pp[lane]), EPSN);
      float na = fmaxf(sqrtf(s_aa[lane]), EPSN);
      v = 1.0f - s_pa[lane] / (np * na);
    }
    v += __shfl_xor(v, 8, 32);
    v += __shfl_xor(v, 4, 32);
    v += __shfl_xor(v, 2, 32);
    v += __shfl_xor(v, 1, 32);
    if (lane == 0) atomicAdd(out, v / (float)N);  // mean(1 - cos)
  }
}

extern "C" void kernel_launch(void* const* d_in, const int* in_sizes, int n_in,
                              void* d_out, int out_size, void* d_ws, size_t ws_size,
                              hipStream_t stream) {
  const float* pc  = (const float*)d_in[0];
  const float* aug = (const float*)d_in[1];
  float* out = (float*)d_out;

  const int D = 1024;                  // feature dim per reference
  const int N = in_sizes[0] / D;       // 16384 rows

  // d_out is poisoned before timing; zero it each call (graph-capturable).
  hipMemsetAsync(out, 0, sizeof(float) * (size_t)out_size, stream);

  dim3 grid(N / TILE_M);               // 1024 blocks
  dim3 block(TILE_M * 2 * WAVES_PER_BLOCK);  // 128 threads = 4 waves
  hipLaunchKernelGGL(CosinLoss_wmma_kernel, grid, block, 0, stream,
                     pc, aug, out, N, D);
}